// KeysPressesLoss_64029372449105
// MI455X (gfx1250) — compile-verified
//
#include <hip/hip_runtime.h>

#define ROWS_PER_BLOCK 256
#define DCOLS 12

// ---------- CDNA5 async global->LDS copy (16 bytes / lane) ----------
#if defined(__has_builtin)
#  if __has_builtin(__builtin_amdgcn_global_load_async_to_lds_b128)
#    define USE_ASYNC_BUILTIN 1
#  endif
#  if __has_builtin(__builtin_amdgcn_s_wait_asynccnt)
#    define USE_WAIT_BUILTIN 1
#  endif
#endif

// Match the builtin's parameter type exactly: vector of 4 ints (16 bytes),
// global (addrspace 1) source, LDS (addrspace 3) destination.
typedef int v4i_vs __attribute__((vector_size(16)));
typedef __attribute__((address_space(1))) v4i_vs* g_v4i_ptr;
typedef __attribute__((address_space(3))) v4i_vs* l_v4i_ptr;

__device__ __forceinline__ void async_copy_b128(const float* g, float* l) {
#if defined(USE_ASYNC_BUILTIN)
    __builtin_amdgcn_global_load_async_to_lds_b128(
        (g_v4i_ptr)(unsigned long long)g,   // drop const + cast to as(1) v4i*
        (l_v4i_ptr)l,
        /*offset=*/0, /*cpol=*/0);
#else
    // Fallback: VDST = LDS byte offset (low 32 bits of generic LDS address),
    // VADDR = 64-bit global address, SADDR = off.
    unsigned lds_off = (unsigned)(unsigned long long)l;
    asm volatile("global_load_async_to_lds_b128 %0, %1, off"
                 :: "v"(lds_off), "v"(g)
                 : "memory");
#endif
}

__device__ __forceinline__ void wait_async_zero() {
#if defined(USE_WAIT_BUILTIN)
    __builtin_amdgcn_s_wait_asynccnt(0);
#else
    asm volatile("s_wait_asynccnt 0" ::: "memory");
#endif
}

// log2 via hardware v_log_f32 (TRANS op, cheap; we're bandwidth-bound anyway)
__device__ __forceinline__ float hlog2(float x) {
    return __builtin_amdgcn_logf(x);
}

__global__ void __launch_bounds__(ROWS_PER_BLOCK)
fused_bce_mse_loss_kernel(const float* __restrict__ y_pred,
                          const float* __restrict__ y_true,
                          float* __restrict__ out,
                          int n_rows) {
    // One 256-row tile per block: 256*12 floats = 12 KB per input in LDS.
    __shared__ float lds_p[ROWS_PER_BLOCK * DCOLS];
    __shared__ float lds_t[ROWS_PER_BLOCK * DCOLS];

    const int t = threadIdx.x;
    const long long row0  = (long long)blockIdx.x * ROWS_PER_BLOCK;
    const long long base  = row0 * DCOLS;                 // first float of tile
    const long long total = (long long)n_rows * DCOLS;    // total floats/input

    // Stage the tile: 3 fully coalesced 16B-per-lane async chunks per input.
    // Per wave, each instruction moves 512 contiguous bytes global -> LDS.
#pragma unroll
    for (int c = 0; c < 3; ++c) {
        const int u = c * ROWS_PER_BLOCK + t;             // 16B chunk id
        const long long e = base + (long long)u * 4;      // float index
        if (e < total) {                                  // total % 4 == 0
            async_copy_b128(y_pred + e, &lds_p[u * 4]);
            async_copy_b128(y_true + e, &lds_t[u * 4]);
        }
    }
    wait_async_zero();   // my wave's async transfers landed in LDS
    __syncthreads();     // everyone's transfers landed

    const long long row = row0 + t;
    if (row < n_rows) {
        // Row base = t*48 bytes -> 16B aligned: three ds_load_b128 per input.
        const float4* p4 = reinterpret_cast<const float4*>(&lds_p[t * DCOLS]);
        const float4* q4 = reinterpret_cast<const float4*>(&lds_t[t * DCOLS]);
        const float4 p0 = p4[0], p1 = p4[1], p2 = p4[2];
        const float4 q0 = q4[0], q1 = q4[1], q2 = q4[2];

        // Columns 0..9: BCE (log2 domain). Columns 10,11: squared error.
        float s = 0.0f;
        s += q0.x * hlog2(p0.x) + (1.0f - q0.x) * hlog2(1.0f - p0.x);
        s += q0.y * hlog2(p0.y) + (1.0f - q0.y) * hlog2(1.0f - p0.y);
        s += q0.z * hlog2(p0.z) + (1.0f - q0.z) * hlog2(1.0f - p0.z);
        s += q0.w * hlog2(p0.w) + (1.0f - q0.w) * hlog2(1.0f - p0.w);
        s += q1.x * hlog2(p1.x) + (1.0f - q1.x) * hlog2(1.0f - p1.x);
        s += q1.y * hlog2(p1.y) + (1.0f - q1.y) * hlog2(1.0f - p1.y);
        s += q1.z * hlog2(p1.z) + (1.0f - q1.z) * hlog2(1.0f - p1.z);
        s += q1.w * hlog2(p1.w) + (1.0f - q1.w) * hlog2(1.0f - p1.w);
        s += q2.x * hlog2(p2.x) + (1.0f - q2.x) * hlog2(1.0f - p2.x);
        s += q2.y * hlog2(p2.y) + (1.0f - q2.y) * hlog2(1.0f - p2.y);

        const float d10 = p2.z - q2.z;
        const float d11 = p2.w - q2.w;
        const float mse = d10 * d10 + d11 * d11;

        const float res = (mse - 0.69314718055994531f * s) * (1.0f / 12.0f);
        __builtin_nontemporal_store(res, out + row);
    }
}

extern "C" void kernel_launch(void* const* d_in, const int* in_sizes, int n_in,
                              void* d_out, int out_size, void* d_ws, size_t ws_size,
                              hipStream_t stream) {
    (void)n_in; (void)d_ws; (void)ws_size; (void)in_sizes;
    const float* y_pred = (const float*)d_in[0];
    const float* y_true = (const float*)d_in[1];
    float* out = (float*)d_out;

    const int n_rows = out_size;  // reference output is [N]
    const int blocks = (n_rows + ROWS_PER_BLOCK - 1) / ROWS_PER_BLOCK;
    if (blocks > 0) {
        fused_bce_mse_loss_kernel<<<dim3(blocks), dim3(ROWS_PER_BLOCK), 0, stream>>>(
            y_pred, y_true, out, n_rows);
    }
}